// MapDensePoseTexModule_84267258347702
// MI455X (gfx1250) — compile-verified
//
#include <hip/hip_runtime.h>

typedef int   v4i __attribute__((ext_vector_type(4)));
typedef float v4f __attribute__((ext_vector_type(4)));
typedef float v2f __attribute__((ext_vector_type(2)));

#define NB   16
#define NC   3
#define TH_  512
#define TW_  512
#define OH_  768
#define OW_  768
#define PIX  (OH_ * OW_)        /* 589824, multiple of 4 */
#define PPT  4                  /* pixels per thread     */
#define TPB  256

__global__ __launch_bounds__(TPB)
void densepose_tex_map(const float* __restrict__ tex,
                       const int*   __restrict__ iuv,
                       const float* __restrict__ lut,
                       float*       __restrict__ out)
{
    const long long gid = (long long)blockIdx.x * TPB + threadIdx.x;
    const long long p0  = gid * PPT;
    if (p0 >= (long long)NB * PIX) return;
    const int b   = (int)(p0 / PIX);
    const int rem = (int)(p0 - (long long)b * PIX);

    const int* iuvb = iuv + (long long)b * 3 * PIX + rem;

    v4i iv, uvl, vvl;
#if __has_builtin(__builtin_amdgcn_global_load_async_to_lds_b128)
    // CDNA5 async global->LDS staging of the streaming iuv planes (ASYNCcnt path).
    // Builtin params are typed v4i pointers: (AS1 global src, AS3 lds dst, imm offset, imm cpol).
    __shared__ int lds_iuv[3][TPB * PPT];
    {
        typedef __attribute__((address_space(1))) v4i gv4i;
        typedef __attribute__((address_space(3))) v4i lv4i;
        const int t4 = threadIdx.x * PPT;
        __builtin_amdgcn_global_load_async_to_lds_b128(
            (gv4i*)(iuvb),           (lv4i*)&lds_iuv[0][t4], 0, 0);
        __builtin_amdgcn_global_load_async_to_lds_b128(
            (gv4i*)(iuvb + PIX),     (lv4i*)&lds_iuv[1][t4], 0, 0);
        __builtin_amdgcn_global_load_async_to_lds_b128(
            (gv4i*)(iuvb + 2 * PIX), (lv4i*)&lds_iuv[2][t4], 0, 0);
#if __has_builtin(__builtin_amdgcn_s_wait_asynccnt)
        __builtin_amdgcn_s_wait_asynccnt(0);
#else
        asm volatile("s_wait_asynccnt 0" ::: "memory");
#endif
        // Each lane reads back only its own 16B per plane: no barrier needed.
        iv  = *(const v4i*)&lds_iuv[0][t4];
        uvl = *(const v4i*)&lds_iuv[1][t4];
        vvl = *(const v4i*)&lds_iuv[2][t4];
    }
#else
    // Fallback: non-temporal B128 loads (read-once stream; keep L2 for tex/lut).
    iv  = __builtin_nontemporal_load((const v4i*)(iuvb));
    uvl = __builtin_nontemporal_load((const v4i*)(iuvb + PIX));
    vvl = __builtin_nontemporal_load((const v4i*)(iuvb + 2 * PIX));
#endif

    const float* texb = tex + (long long)b * NC * TH_ * TW_;
    const float* t0 = texb;
    const float* t1 = texb + TH_ * TW_;
    const float* t2 = texb + 2 * TH_ * TW_;

    v4f r0, r1, r2;
#pragma unroll
    for (int j = 0; j < PPT; ++j) {
        const int ii = iv[j];
        int ui = uvl[j]; ui = ui < 0 ? 0 : (ui > 255 ? 255 : ui);   // == round(clip(u/255)*255)
        int vi = vvl[j]; vi = vi < 0 ? 0 : (vi > 255 ? 255 : vi);

        const float* lp = lut + (((long long)ii * 256 + (long long)vi) * 256 + ui) * 2;
        const v2f luv = *(const v2f*)lp;                            // 8B aligned gather, L2-hot

        // grid math folded: x = ((u_I+1)/2)*(W-1) = luv.x*511 ; y = (1-luv.y)*511
        const float x = luv.x * (float)(TW_ - 1);
        const float y = (1.0f - luv.y) * (float)(TH_ - 1);
        const float x0f = floorf(x), y0f = floorf(y);
        const float wx = x - x0f,    wy = y - y0f;
        const int x0 = (int)x0f, y0 = (int)y0f;
        const int x1 = x0 + 1,   y1 = y0 + 1;

        const bool vx0 = (unsigned)x0 < (unsigned)TW_;
        const bool vx1 = (unsigned)x1 < (unsigned)TW_;
        const bool vy0 = (unsigned)y0 < (unsigned)TH_;
        const bool vy1 = (unsigned)y1 < (unsigned)TH_;

        float w00 = (1.0f - wy) * (1.0f - wx); if (!(vy0 && vx0)) w00 = 0.0f;
        float w01 = (1.0f - wy) * wx;          if (!(vy0 && vx1)) w01 = 0.0f;
        float w10 = wy * (1.0f - wx);          if (!(vy1 && vx0)) w10 = 0.0f;
        float w11 = wy * wx;                   if (!(vy1 && vx1)) w11 = 0.0f;

        const int xc0 = x0 < 0 ? 0 : (x0 > TW_ - 1 ? TW_ - 1 : x0);
        const int xc1 = x1 < 0 ? 0 : (x1 > TW_ - 1 ? TW_ - 1 : x1);
        const int yc0 = y0 < 0 ? 0 : (y0 > TH_ - 1 ? TH_ - 1 : y0);
        const int yc1 = y1 < 0 ? 0 : (y1 > TH_ - 1 ? TH_ - 1 : y1);

        const int o00 = yc0 * TW_ + xc0;
        const int o01 = yc0 * TW_ + xc1;
        const int o10 = yc1 * TW_ + xc0;
        const int o11 = yc1 * TW_ + xc1;

        r0[j] = t0[o00] * w00 + t0[o01] * w01 + t0[o10] * w10 + t0[o11] * w11;
        r1[j] = t1[o00] * w00 + t1[o01] * w01 + t1[o10] * w10 + t1[o11] * w11;
        r2[j] = t2[o00] * w00 + t2[o01] * w01 + t2[o10] * w10 + t2[o11] * w11;
    }

    // Write-once output streams: non-temporal B128 stores, one per channel plane.
    float* ob = out + (long long)b * NC * PIX + rem;
    __builtin_nontemporal_store(r0, (v4f*)(ob));
    __builtin_nontemporal_store(r1, (v4f*)(ob + PIX));
    __builtin_nontemporal_store(r2, (v4f*)(ob + 2 * PIX));
}

extern "C" void kernel_launch(void* const* d_in, const int* in_sizes, int n_in,
                              void* d_out, int out_size, void* d_ws, size_t ws_size,
                              hipStream_t stream)
{
    (void)in_sizes; (void)n_in; (void)out_size; (void)d_ws; (void)ws_size;
    const float* tex = (const float*)d_in[0];   // [16,3,512,512] f32
    const int*   iuv = (const int*)  d_in[1];   // [16,3,768,768] i32
    const float* lut = (const float*)d_in[2];   // [25,256,256,2] f32
    float*       out = (float*)d_out;           // [16,3,768,768] f32

    const long long total_threads = (long long)NB * PIX / PPT;      // 2,359,296
    const int blocks = (int)((total_threads + TPB - 1) / TPB);      // 9,216 — exact
    densepose_tex_map<<<blocks, TPB, 0, stream>>>(tex, iuv, lut, out);
}